// EfficientLinearProbe_39393440039021
// MI455X (gfx1250) — compile-verified
//
#include <hip/hip_runtime.h>
#include <hip/hip_bf16.h>

typedef float v2f __attribute__((ext_vector_type(2)));
typedef float v4f __attribute__((ext_vector_type(4)));
typedef float v8f __attribute__((ext_vector_type(8)));

// Problem shape (fixed by reference setup_inputs): B=16, S=2048, H=512.
constexpr int Bc = 16;
constexpr int Sc = 2048;
constexpr int Hc = 512;

// ---------------------------------------------------------------------------
// Phase 1: u[b,s] = 0.5 * dot(emb[b,s,:], w_l + w_r) + 0.5*bias
// One wave computes 16 rows via V_WMMA_F32_16X16X4_F32.
//   A (16x4 f32):  lanes 0-15 hold row M=lane, K = k,k+1 (VGPR0,VGPR1)
//                  lanes 16-31 hold row M=lane-16, K = k+2,k+3
//   B (4x16 f32):  B[k][n] = 0.5*(w_l[k]+w_r[k]) for ALL n  -> every column
//                  of D equals the dot product. Same per-lane K indexing as A.
//   D (16x16 f32): VGPR v: lanes 0-15 -> M=v, lanes 16-31 -> M=v+8.
//                  Lane 0 / lane 16 each hold 8 row results in their 8 VGPRs.
// ---------------------------------------------------------------------------
__global__ __launch_bounds__(256) void probe_dot_wmma(
    const float* __restrict__ E,     // (B*S, H) row-major
    const float* __restrict__ W,     // (2H,)
    const float* __restrict__ bias,  // (1,)
    float* __restrict__ t)           // (B*S,) output
{
    const int lane = threadIdx.x & 31;
    const int wave = threadIdx.x >> 5;
    const int tile = blockIdx.x * 8 + wave;        // 16 rows per tile
    const long r0  = (long)tile * 16;

    const int m    = lane & 15;
    const int koff = (lane >> 4) << 1;             // 0 for lanes 0-15, 2 for 16-31

    const float* rowp = E + (r0 + m) * Hc + koff;
    const float* wp   = W + koff;

    v8f acc0 = {};
    v8f acc1 = {};

    // 512 K-values, 4 per WMMA, two independent chains -> 64 iterations x 2 WMMA
    for (int k = 0; k < Hc; k += 8) {
        v2f a0, a1, b0, b1;
        a0.x = rowp[k];
        a0.y = rowp[k + 1];
        a1.x = rowp[k + 4];
        a1.y = rowp[k + 5];
        b0.x = 0.5f * (wp[k]     + wp[k + Hc]);
        b0.y = 0.5f * (wp[k + 1] + wp[k + 1 + Hc]);
        b1.x = 0.5f * (wp[k + 4] + wp[k + 4 + Hc]);
        b1.y = 0.5f * (wp[k + 5] + wp[k + 5 + Hc]);
        acc0 = __builtin_amdgcn_wmma_f32_16x16x4_f32(
            false, a0, false, b0, (short)0, acc0, false, false);
        acc1 = __builtin_amdgcn_wmma_f32_16x16x4_f32(
            false, a1, false, b1, (short)0, acc1, false, false);
    }

    v8f acc = acc0 + acc1;
    const float hb = 0.5f * bias[0];

    // lane 0 holds rows r0+0..7 (N=0 column), lane 16 holds rows r0+8..15
    if ((lane & 15) == 0) {
        const long base = r0 + (lane >> 4) * 8;
        #pragma unroll
        for (int v = 0; v < 8; ++v) {
            t[base + v] = acc[v] + hb;
        }
    }
}

// ---------------------------------------------------------------------------
// Phase 2: logits[b,i,j] = (|i-j| >= min_dist) ? u[b,i]+u[b,j] : -inf
// One float4 (along j) per thread; non-temporal 128-bit streaming stores.
// ---------------------------------------------------------------------------
__global__ __launch_bounds__(256) void fill_logits(
    const float* __restrict__ t,
    const int* __restrict__ mdp,
    float* __restrict__ out)
{
    const int md = mdp[0];
    const long gid = (long)blockIdx.x * blockDim.x + threadIdx.x; // B*S*S/4 threads

    const int  j4   = (int)(gid & (Sc / 4 - 1));   // which float4 along j
    const long rest = gid >> 9;                    // / (S/4)
    const int  i    = (int)(rest & (Sc - 1));
    const int  b    = (int)(rest >> 11);           // / S
    const int  j0   = j4 << 2;

    const float ti = t[(long)b * Sc + i];
    const v4f   tj = *(const v4f*)(t + (long)b * Sc + j0);

    const float ninf = -__builtin_inff();
    v4f outv;
    #pragma unroll
    for (int c = 0; c < 4; ++c) {
        const int j = j0 + c;
        int d = i - j;
        if (d < 0) d = -d;
        outv[c] = (d >= md) ? (ti + tj[c]) : ninf;
    }
    __builtin_nontemporal_store(outv, (v4f*)out + gid);
}

// ---------------------------------------------------------------------------
// Phase 3: mask[i,j] = |i-j| >= min_dist, as 0.0/1.0 floats, streamed.
// ---------------------------------------------------------------------------
__global__ __launch_bounds__(256) void fill_mask(
    const int* __restrict__ mdp,
    float* __restrict__ out)
{
    const int md  = mdp[0];
    const int gid = blockIdx.x * blockDim.x + threadIdx.x;  // S*S/4 threads

    const int j4 = gid & (Sc / 4 - 1);
    const int i  = gid >> 9;
    const int j0 = j4 << 2;

    v4f v;
    #pragma unroll
    for (int c = 0; c < 4; ++c) {
        int d = i - (j0 + c);
        if (d < 0) d = -d;
        v[c] = (d >= md) ? 1.0f : 0.0f;
    }
    __builtin_nontemporal_store(v, (v4f*)out + gid);
}

extern "C" void kernel_launch(void* const* d_in, const int* in_sizes, int n_in,
                              void* d_out, int out_size, void* d_ws, size_t ws_size,
                              hipStream_t stream) {
    const float* E    = (const float*)d_in[0];  // (B,S,H) f32
    const float* W    = (const float*)d_in[1];  // (2H,)   f32
    const float* bias = (const float*)d_in[2];  // (1,)    f32
    const int*   md   = (const int*)d_in[3];    // scalar int

    float* out = (float*)d_out;
    float* t   = (float*)d_ws;                  // B*S floats = 128 KB scratch

    // Phase 1: (B*S)/16 = 2048 wave-tiles, 8 waves per 256-thread block.
    probe_dot_wmma<<<(Bc * Sc) / (16 * 8), 256, 0, stream>>>(E, W, bias, t);

    // Phase 2: B*S*S/4 float4 stores.
    const long n_vec4 = (long)Bc * Sc * Sc / 4;
    fill_logits<<<(int)(n_vec4 / 256), 256, 0, stream>>>(t, md, out);

    // Phase 3: mask appended after logits.
    const int n_mask4 = Sc * Sc / 4;
    fill_mask<<<n_mask4 / 256, 256, 0, stream>>>(md, out + (long)Bc * Sc * Sc);
}